// BilateralDenoiser_27453430956362
// MI455X (gfx1250) — compile-verified
//
#include <hip/hip_runtime.h>
#include <stdint.h>

// Bilateral denoiser, CDNA5/gfx1250.
// sigma=2 -> rad=5 -> 11x11 window; normal power 128; eps 1e-4.
#define RAD   5
#define TS    16
#define HT    (TS + 2*RAD)     // 26 halo tile side
#define NPX   (HT * HT)        // 676 halo pixels
#define NTHR  (TS * TS)        // 256 threads = 8 waves (wave32)

#define L2E      1.4426950408889634f          // log2(e)
#define T2MAX    (10000.0f * L2E)             // (1/eps) * log2e
#define NEG_I2V  (-0.125f)                    // -1/(2*sigma^2)

// ---- CDNA5 async global->LDS DMA (ASYNCcnt path) -------------------------
__device__ __forceinline__ void async_b32_to_lds(uint32_t lds_byte_off, const void* g) {
    // GLOBAL_LOAD_ASYNC_TO_LDS_B32 (GV mode): per-lane LDS dest VGPR, 64-bit vaddr.
    asm volatile("global_load_async_to_lds_b32 %0, %1, off"
                 :: "v"(lds_byte_off),
                    "v"((unsigned long long)(uintptr_t)g)
                 : "memory");
}

__device__ __forceinline__ void wait_asynccnt0() {
#if __has_builtin(__builtin_amdgcn_s_wait_asynccnt)
    __builtin_amdgcn_s_wait_asynccnt(0);
#else
    asm volatile("s_wait_asynccnt 0" ::: "memory");
#endif
}

__device__ __forceinline__ float fast_rcp(float x) {
#if __has_builtin(__builtin_amdgcn_rcpf)
    return __builtin_amdgcn_rcpf(x);
#else
    return 1.0f / x;
#endif
}

__device__ __forceinline__ float fast_rsq(float x) {
#if __has_builtin(__builtin_amdgcn_rsqf)
    return __builtin_amdgcn_rsqf(x);
#else
    return 1.0f / __builtin_sqrtf(x);
#endif
}

__device__ __forceinline__ float fast_exp2(float x) {
#if __has_builtin(__builtin_amdgcn_exp2f)
    return __builtin_amdgcn_exp2f(x);
#else
    return __builtin_exp2f(x);
#endif
}

__device__ __forceinline__ float clamp01(float x) {
#if __has_builtin(__builtin_amdgcn_fmed3f)
    return __builtin_amdgcn_fmed3f(x, 0.0f, 1.0f);   // v_med3_f32
#else
    return fminf(fmaxf(x, 0.0f), 1.0f);
#endif
}

__global__ __launch_bounds__(NTHR)
void BilateralDenoiser_kernel(const float* __restrict__ col,
                              const float* __restrict__ nrm,
                              const float* __restrict__ zdz,
                              float* __restrict__ out,
                              int H, int W)
{
    // Two separate arrays of 16B/pixel records so each half-wave b128 read
    // covers all 64 LDS banks exactly once (16B lane stride, conflict-free).
    __shared__ float s_nz[NPX * 4];   // {nx, ny, nz, z}
    __shared__ float s_c4[NPX * 4];   // {r, g, b, unused}

    const int b  = blockIdx.z;
    const int x0 = blockIdx.x * TS;
    const int y0 = blockIdx.y * TS;
    const int tx = threadIdx.x, ty = threadIdx.y;
    const int tid = ty * TS + tx;

    const size_t imgPx = (size_t)H * W;
    const float* colB = col + (size_t)b * imgPx * 3;
    const float* nrmB = nrm + (size_t)b * imgPx * 3;
    const float* zdzB = zdz + (size_t)b * imgPx * 2;

    // Low 32 bits of a generic pointer to LDS == byte offset within LDS alloc.
    const uint32_t lds_nz = (uint32_t)(uintptr_t)s_nz;
    const uint32_t lds_c4 = (uint32_t)(uintptr_t)s_c4;

    // ---- Stage halo tile into LDS via async DMA (skip OOB; never read) ----
    for (int i = tid; i < NPX * 3; i += NTHR) {
        int p  = i / 3;                  // halo pixel index
        int c  = i - p * 3;              // channel
        int r  = p / HT;                 // halo row
        int px = p - r * HT;             // halo col
        int gy = y0 - RAD + r;
        int gx = x0 - RAD + px;
        if (gy >= 0 && gy < H && gx >= 0 && gx < W) {
            size_t goff = ((size_t)gy * W + gx) * 3 + c;
            uint32_t rec = (uint32_t)(p * 4 + c) * 4;    // slot c of 16B record
            async_b32_to_lds(lds_c4 + rec, colB + goff);
            async_b32_to_lds(lds_nz + rec, nrmB + goff);
        }
    }
    for (int i = tid; i < NPX; i += NTHR) {
        int r  = i / HT;
        int cx = i - r * HT;
        int gy = y0 - RAD + r;
        int gx = x0 - RAD + cx;
        if (gy >= 0 && gy < H && gx >= 0 && gx < W)
            async_b32_to_lds(lds_nz + (uint32_t)(i * 4 + 3) * 4,
                             zdzB + ((size_t)gy * W + gx) * 2);   // depth lane
    }
    wait_asynccnt0();
    __syncthreads();

    // ---- Per-pixel bilateral accumulation ---------------------------------
    const int x = x0 + tx, y = y0 + ty;

    const int lc4 = ((ty + RAD) * HT + (tx + RAD)) * 4;
    const float cnx = s_nz[lc4], cny = s_nz[lc4 + 1], cnz = s_nz[lc4 + 2];
    const float cz  = s_nz[lc4 + 3];
    const float cdz = zdzB[((size_t)y * W + x) * 2 + 1];

    // 1/max(cdz*dist, eps) == min(rcp(cdz) * (1/dist), 1/eps)  (all terms >= 0;
    // inf cases clamped by the min; center tap has |dz|==0 so its term is 0).
    const float rcp_cdz = fast_rcp(cdz);

    float ar = 0.f, ag = 0.f, ab_ = 0.f, aw = 0.f;

    // c2       = -d2/(2*sigma^2) * log2e          (literal on interior path)
    // dinv_l2e = log2e / dist                     (literal on interior path)
    auto tap = [&](int lr, int lx, float c2, float dinv_l2e) {
        const int q = (lr * HT + lx) * 4;
        const float tnx = s_nz[q],     tny = s_nz[q + 1];   // ds_load_b128
        const float tnz = s_nz[q + 2], tz  = s_nz[q + 3];

        float dot = tnx * cnx + tny * cny + tnz * cnz;
        dot = clamp01(dot);
        float p = dot * dot;                               // ^2
        p *= p; p *= p; p *= p; p *= p; p *= p; p *= p;    // -> ^128

        const float t2  = fminf(rcp_cdz * dinv_l2e, T2MAX);
        const float arg = c2 - fabsf(tz - cz) * t2;        // fma w/ |.| modifier
        const float w   = fast_exp2(arg) * p;              // one v_exp_f32 / tap

        ar  += s_c4[q]     * w;                            // ds_load_b128
        ag  += s_c4[q + 1] * w;
        ab_ += s_c4[q + 2] * w;
        aw  += w;
    };

    const bool interior = (y >= RAD) & (y < H - RAD) & (x >= RAD) & (x < W - RAD);
    if (interior) {
#pragma unroll
        for (int dy = -RAD; dy <= RAD; ++dy) {
#pragma unroll
            for (int dx = -RAD; dx <= RAD; ++dx) {
                const float d2 = (float)(dx * dx + dy * dy);
                const float di = (d2 > 0.f) ? (L2E / __builtin_sqrtf(d2))
                                            : __builtin_inff();
                tap(ty + RAD + dy, tx + RAD + dx, d2 * (NEG_I2V * L2E), di);
            }
        }
    } else {
        const int dylo = max(-RAD, -y), dyhi = min(RAD, H - 1 - y);
        const int dxlo = max(-RAD, -x), dxhi = min(RAD, W - 1 - x);
        for (int dy = dylo; dy <= dyhi; ++dy)
            for (int dx = dxlo; dx <= dxhi; ++dx) {
                const float d2 = (float)(dx * dx + dy * dy);
                tap(ty + RAD + dy, tx + RAD + dx,
                    d2 * (NEG_I2V * L2E), fast_rsq(d2) * L2E);
            }
    }

    const float inv = fast_rcp(aw);     // center tap guarantees aw >= ~1
    const size_t o = ((size_t)b * imgPx + (size_t)y * W + x) * 3;
    out[o]     = ar  * inv;
    out[o + 1] = ag  * inv;
    out[o + 2] = ab_ * inv;
}

extern "C" void kernel_launch(void* const* d_in, const int* in_sizes, int n_in,
                              void* d_out, int out_size, void* d_ws, size_t ws_size,
                              hipStream_t stream) {
    const float* col = (const float*)d_in[0];
    const float* nrm = (const float*)d_in[1];
    const float* zdz = (const float*)d_in[2];
    float* out = (float*)d_out;

    const int H = 1024, W = 1024;
    const int B = in_sizes[0] / (H * W * 3);

    dim3 grid(W / TS, H / TS, B);
    dim3 block(TS, TS, 1);
    hipLaunchKernelGGL(BilateralDenoiser_kernel, grid, block, 0, stream,
                       col, nrm, zdz, out, H, W);
}